// HcSplitSinkhorn_41798621724793
// MI455X (gfx1250) — compile-verified
//
#include <hip/hip_runtime.h>

typedef __attribute__((ext_vector_type(4))) unsigned int v4u;
typedef __attribute__((ext_vector_type(8))) int          v8i;
typedef __attribute__((ext_vector_type(4))) int          v4i;

#define EPS_F      1e-6f
#define LOG2E_F    1.44269504088896340736f
#define TOK_PER_BLK 256
#define FLT_PER_TOK 24   // 4 pre + 4 post + 16 comb

__device__ __forceinline__ float fast_rcp(float x)  { return __builtin_amdgcn_rcpf(x); }
__device__ __forceinline__ float fast_exp(float x)  { return __builtin_amdgcn_exp2f(x * LOG2E_F); }
__device__ __forceinline__ float fast_sigmoid(float x) { return fast_rcp(1.0f + fast_exp(-x)); }

__device__ __forceinline__ void row_norm(float (&m)[4][4]) {
  #pragma unroll
  for (int r = 0; r < 4; ++r) {
    float inv = fast_rcp(m[r][0] + m[r][1] + m[r][2] + m[r][3] + EPS_F);
    m[r][0] *= inv; m[r][1] *= inv; m[r][2] *= inv; m[r][3] *= inv;
  }
}
__device__ __forceinline__ void col_norm(float (&m)[4][4]) {
  #pragma unroll
  for (int c = 0; c < 4; ++c) {
    float inv = fast_rcp(m[0][c] + m[1][c] + m[2][c] + m[3][c] + EPS_F);
    m[0][c] *= inv; m[1][c] *= inv; m[2][c] *= inv; m[3][c] *= inv;
  }
}

__global__ __launch_bounds__(TOK_PER_BLK)
void HcSplitSinkhorn_kernel(const float* __restrict__ mixes,
                            const float* __restrict__ hc_scale,
                            const float* __restrict__ hc_base,
                            float* __restrict__ out,
                            int n_tokens) {
  __shared__ __align__(16) float tile[TOK_PER_BLK * FLT_PER_TOK];

  const int tid  = threadIdx.x;
  const int tok0 = blockIdx.x * TOK_PER_BLK;
  int rem = n_tokens - tok0;
  if (rem > TOK_PER_BLK) rem = TOK_PER_BLK;
  const unsigned n_elem = (unsigned)(rem * FLT_PER_TOK);   // elements this block stages

  // ---- TDM: DMA this block's 256x24 float record slab into LDS (one descriptor) ----
  if (tid == 0) {
    unsigned lds_addr = (unsigned)(unsigned long long)(&tile[0]);   // low 32 bits = LDS offset
    unsigned long long ga =
        (unsigned long long)(const void*)(mixes + (size_t)tok0 * FLT_PER_TOK);
    // D# group 0: count=1 | lds_addr | global_addr[31:0] | global_addr[56:32], type=2
    v4u g0 = { 1u,
               lds_addr,
               (unsigned)ga,
               (unsigned)((ga >> 32) & 0x01FFFFFFull) | (2u << 30) };
    // D# group 1: data_size=4B; tensor_dim0 = tile_dim0 = n_elem (1-D tile);
    //             tensor_dim1 = 1; tile_dim1 = tile_dim2 = 1; strides = n_elem
    v8i g1 = { (int)(2u << 16),                       // [17:16] data_size = 2 -> 4 bytes
               (int)(n_elem << 16),                   // [47:32] bar addr=0, [63:48] dim0 lo16
               (int)((n_elem >> 16) | (1u << 16)),    // [79:64] dim0 hi16, [95:80] dim1 lo16 = 1
               (int)(n_elem << 16),                   // [111:96] dim1 hi16 = 0, [127:112] tile_dim0
               (int)(1u | (1u << 16)),                // tile_dim1 = 1, tile_dim2 = 1
               (int)n_elem,                           // tensor_dim0_stride lo32
               (int)(n_elem << 16),                   // stride0 hi16=0, tensor_dim1_stride lo16
               0 };                                   // tensor_dim1_stride hi32
    v4i g2 = { 1, 1, (int)n_elem, 0 };                // tensor_dim2=1, tensor_dim3=1, dim2_stride
    v4i g3 = { (int)n_elem, (int)(1u << 16), 0, 0 };  // dim3_stride, tensor_dim4=1, tile_dim4=0
    v8i g4 = { 0, 0, 0, 0, 0, 0, 0, 0 };              // extra group (unused for 1-D tile)
    __builtin_amdgcn_tensor_load_to_lds(g0, g1, g2, g3, g4, 0);
    __builtin_amdgcn_s_wait_tensorcnt(0);
  }
  __syncthreads();

  if (tid >= rem) return;

  const float s0 = hc_scale[0], s1 = hc_scale[1], s2 = hc_scale[2];

  // ---- pull this token's 24-float record out of LDS (6x ds_load_b128) ----
  float v[FLT_PER_TOK];
  const float* my = &tile[tid * FLT_PER_TOK];
  #pragma unroll
  for (int i = 0; i < 6; ++i) {
    float4 t = *(const float4*)(my + 4 * i);
    v[4*i + 0] = t.x; v[4*i + 1] = t.y; v[4*i + 2] = t.z; v[4*i + 3] = t.w;
  }

  // ---- pre / post gates ----
  float pre[4], post[4];
  #pragma unroll
  for (int i = 0; i < 4; ++i)
    pre[i] = fast_sigmoid(v[i] * s0 + hc_base[i]) + EPS_F;
  #pragma unroll
  for (int i = 0; i < 4; ++i)
    post[i] = 2.0f * fast_sigmoid(v[4 + i] * s1 + hc_base[4 + i]);

  // ---- comb: affine + row softmax (+EPS) ----
  float m[4][4];
  #pragma unroll
  for (int r = 0; r < 4; ++r)
    #pragma unroll
    for (int c = 0; c < 4; ++c)
      m[r][c] = v[8 + 4*r + c] * s2 + hc_base[8 + 4*r + c];

  #pragma unroll
  for (int r = 0; r < 4; ++r) {
    float mx = fmaxf(fmaxf(m[r][0], m[r][1]), fmaxf(m[r][2], m[r][3]));
    float e0 = fast_exp(m[r][0] - mx);
    float e1 = fast_exp(m[r][1] - mx);
    float e2 = fast_exp(m[r][2] - mx);
    float e3 = fast_exp(m[r][3] - mx);
    float inv = fast_rcp(e0 + e1 + e2 + e3);
    m[r][0] = e0 * inv + EPS_F;
    m[r][1] = e1 * inv + EPS_F;
    m[r][2] = e2 * inv + EPS_F;
    m[r][3] = e3 * inv + EPS_F;
  }

  // ---- Sinkhorn: col-norm, then 19 x (row-norm, col-norm) ----
  col_norm(m);
  #pragma unroll 1
  for (int it = 0; it < 19; ++it) {
    row_norm(m);
    col_norm(m);
  }

  // ---- coalesced float4 stores: [pre | post | comb] flat ----
  const size_t tok = (size_t)tok0 + (size_t)tid;
  const size_t NT  = (size_t)n_tokens;
  float4* preO  = (float4*)(out + tok * 4);
  float4* postO = (float4*)(out + NT * 4 + tok * 4);
  float4* combO = (float4*)(out + NT * 8 + tok * 16);
  *preO  = make_float4(pre[0],  pre[1],  pre[2],  pre[3]);
  *postO = make_float4(post[0], post[1], post[2], post[3]);
  #pragma unroll
  for (int r = 0; r < 4; ++r)
    combO[r] = make_float4(m[r][0], m[r][1], m[r][2], m[r][3]);
}

extern "C" void kernel_launch(void* const* d_in, const int* in_sizes, int n_in,
                              void* d_out, int out_size, void* d_ws, size_t ws_size,
                              hipStream_t stream) {
  const float* mixes    = (const float*)d_in[0];
  const float* hc_scale = (const float*)d_in[1];
  const float* hc_base  = (const float*)d_in[2];
  float* out = (float*)d_out;

  const int n_tokens = in_sizes[0] / FLT_PER_TOK;   // (B*S)
  const int blocks   = (n_tokens + TOK_PER_BLK - 1) / TOK_PER_BLK;

  HcSplitSinkhorn_kernel<<<blocks, TOK_PER_BLK, 0, stream>>>(
      mixes, hc_scale, hc_base, out, n_tokens);
}